// GraphAutoencoder_24472723652810
// MI455X (gfx1250) — compile-verified
//
#include <hip/hip_runtime.h>
#include <math.h>

typedef __attribute__((ext_vector_type(2))) float v2f;
typedef __attribute__((ext_vector_type(8))) float v8f;

#define DH 128        // feature dim (all layers are 128 wide)
#define PSTRIDE 144   // K-pair row stride in float2 units: 144*8B = 1152B == 128 mod 256B
                      // -> lanes 16-31 (next K-pair row) land 32 banks away from lanes 0-15

// ---------------- degree / normalization ----------------

__global__ __launch_bounds__(256) void k_zero(float* __restrict__ p, int n) {
  int i = blockIdx.x * blockDim.x + threadIdx.x;
  if (i < n) p[i] = 0.0f;
}

__global__ __launch_bounds__(256) void k_count(const int* __restrict__ dst,
                                               float* __restrict__ deg, int E) {
  int i = blockIdx.x * blockDim.x + threadIdx.x;
  if (i < E) atomicAdd(&deg[dst[i]], 1.0f);
}

__global__ __launch_bounds__(256) void k_dinv(float* __restrict__ d, int n) {
  int i = blockIdx.x * blockDim.x + threadIdx.x;
  if (i < n) d[i] = rsqrtf(d[i] + 1.0f);  // +1 self loop; deg >= 1
}

// ---------------- H = X @ W via V_WMMA_F32_16X16X4_F32 ----------------
// X: [N][128] row-major, W: [128][128] row-major (K x N), H: [N][128].
// W is staged in LDS K-pair interleaved: sW2[p*PSTRIDE + n] = {W[2p][n], W[2p+1][n]}
// so every B fragment is ONE contiguous 8-byte ds_load_b64 straight into an
// even-aligned VGPR pair (no mov shuffling before the WMMA).
// 256 threads = 8 waves; each wave computes a 16-row strip x all 128 cols.
__global__ __launch_bounds__(256) void k_gemm(const float* __restrict__ X,
                                              const float* __restrict__ W,
                                              float* __restrict__ H, int N) {
  __shared__ v2f sW2[(DH / 2) * PSTRIDE];
  int tid = threadIdx.x;
  for (int i = tid; i < (DH / 2) * DH; i += 256) {
    int p = i >> 7;          // K-pair index 0..63
    int n = i & 127;         // output column
    v2f w;
    w[0] = W[(2 * p) * DH + n];
    w[1] = W[(2 * p + 1) * DH + n];
    sW2[p * PSTRIDE + n] = w;
  }
  __syncthreads();

  int wave = tid >> 5;
  int lane = tid & 31;
  long rowBase = ((long)blockIdx.x * 8 + wave) * 16;
  if (rowBase >= N) return;              // wave-uniform: EXEC stays all-ones for WMMA

  int m     = lane & 15;                 // A: row M within tile; B/D: column N
  int kh    = lane >> 4;                 // 0: lanes hold K,K+1; 1: lanes hold K+2,K+3

  const float* __restrict__ Xrow = X + (rowBase + m) * DH;

  v8f acc[8];
#pragma unroll
  for (int t = 0; t < 8; ++t) acc[t] = (v8f){};

  for (int k = 0; k < DH; k += 4) {
    // A fragment 16x4 f32: VGPR0 = A[M][k+2*kh], VGPR1 = A[M][k+2*kh+1]
    v2f a = *(const v2f*)(Xrow + k + 2 * kh);
    const v2f* __restrict__ wrow = sW2 + (k / 2 + kh) * PSTRIDE + m;
#pragma unroll
    for (int t = 0; t < 8; ++t) {
      // B fragment 4x16 f32 in one ds_load_b64: {B[k+2*kh][n], B[k+2*kh+1][n]}
      v2f b = wrow[t * 16];
      acc[t] = __builtin_amdgcn_wmma_f32_16x16x4_f32(
          /*neg_a=*/false, a, /*neg_b=*/false, b,
          /*c_mod=*/(short)0, acc[t], /*reuse_a=*/false, /*reuse_b=*/false);
    }
  }

  // D layout: VGPR r -> M=r (lanes 0-15) / M=8+r (lanes 16-31); N = lane&15
  int n    = lane & 15;
  int mrow = kh * 8;
#pragma unroll
  for (int t = 0; t < 8; ++t)
#pragma unroll
    for (int r = 0; r < 8; ++r)
      H[(rowBase + mrow + r) * DH + t * 16 + n] = acc[t][r];
}

// ---------------- acc = H * dinv^2 + b (also serves as the zero-init) ----------------

__global__ __launch_bounds__(256) void k_selfbias(const float* __restrict__ H,
                                                  const float* __restrict__ dinv,
                                                  const float* __restrict__ b,
                                                  float* __restrict__ acc, int total) {
  int i = blockIdx.x * blockDim.x + threadIdx.x;
  if (i < total) {
    int row = i >> 7, col = i & 127;
    float di = dinv[row];
    acc[i] = H[i] * di * di + b[col];
  }
}

// ---------------- edge scatter: acc[dst] += H[src] * dinv[src]*dinv[dst] ----------------
// One lane per (edge, 4-feature chunk): 32 lanes cover one edge's 128 features.
__global__ __launch_bounds__(256) void k_scatter(const float* __restrict__ H,
                                                 const int* __restrict__ src,
                                                 const int* __restrict__ dst,
                                                 const float* __restrict__ dinv,
                                                 float* __restrict__ acc, int E) {
  long gid = (long)blockIdx.x * blockDim.x + threadIdx.x;
  if (gid >= (long)E * 32) return;
  int e  = (int)(gid >> 5);
  int c4 = ((int)gid & 31) * 4;
  int s = src[e], d = dst[e];
  float coef = dinv[s] * dinv[d];
  const float4 hv = *(const float4*)(H + (long)s * DH + c4);
  float* p = acc + (long)d * DH + c4;
  atomicAdd(p + 0, hv.x * coef);
  atomicAdd(p + 1, hv.y * coef);
  atomicAdd(p + 2, hv.z * coef);
  atomicAdd(p + 3, hv.w * coef);
}

// ---------------- activation ----------------

__global__ __launch_bounds__(256) void k_act(const float* __restrict__ acc,
                                             float* __restrict__ out, int n, int relu) {
  int i = blockIdx.x * blockDim.x + threadIdx.x;
  if (i < n) {
    float v = acc[i];
    out[i] = relu ? fmaxf(v, 0.0f) : (v > 0.0f ? v : expm1f(v));
  }
}

// ---------------- host-side orchestration ----------------

extern "C" void kernel_launch(void* const* d_in, const int* in_sizes, int n_in,
                              void* d_out, int out_size, void* d_ws, size_t ws_size,
                              hipStream_t stream) {
  const float* x  = (const float*)d_in[0];
  const float* W1 = (const float*)d_in[1];
  const float* b1 = (const float*)d_in[2];
  const float* W2 = (const float*)d_in[3];
  const float* b2 = (const float*)d_in[4];
  const float* W3 = (const float*)d_in[5];
  const float* b3 = (const float*)d_in[6];
  const float* W4 = (const float*)d_in[7];
  const float* b4 = (const float*)d_in[8];
  const int* ei1  = (const int*)d_in[9];   // [2, E1]: src row then dst row
  const int* ei2  = (const int*)d_in[10];

  const int N  = in_sizes[0] / DH;   // 50000 (multiple of 16)
  const int E1 = in_sizes[9] / 2;
  const int E2 = in_sizes[10] / 2;
  const int total = N * DH;

  float* ws    = (float*)d_ws;
  float* h     = ws;                          // [N*128] GEMM result of current layer
  float* acc   = h   + (size_t)total;         // [N*128] scatter accumulator
  float* a     = acc + (size_t)total;         // [N*128] intermediate activation
  float* dinv1 = a   + (size_t)total;         // [N]
  float* dinv2 = dinv1 + (size_t)N;           // [N]

  float* zout  = (float*)d_out;               // first output (after layer 2)
  float* xrout = zout + (size_t)total;        // second output (after layer 4)

  const unsigned gN = (N + 255) / 256;
  const unsigned gF = (total + 255) / 256;
  const unsigned gG = (unsigned)((N + 127) / 128);  // gemm: 128 rows per block

  // dinv for both edge sets (deg computed in place)
  k_zero<<<gN, 256, 0, stream>>>(dinv1, N);
  k_zero<<<gN, 256, 0, stream>>>(dinv2, N);
  k_count<<<(E1 + 255) / 256, 256, 0, stream>>>(ei1 + E1, dinv1, E1);
  k_count<<<(E2 + 255) / 256, 256, 0, stream>>>(ei2 + E2, dinv2, E2);
  k_dinv<<<gN, 256, 0, stream>>>(dinv1, N);
  k_dinv<<<gN, 256, 0, stream>>>(dinv2, N);

  auto layer = [&](const float* in, const float* W, const float* bias,
                   const int* ei, int E, const float* dinv, float* out, int relu) {
    k_gemm<<<gG, 256, 0, stream>>>(in, W, h, N);
    k_selfbias<<<gF, 256, 0, stream>>>(h, dinv, bias, acc, total);
    long tw = (long)E * 32;
    k_scatter<<<(unsigned)((tw + 255) / 256), 256, 0, stream>>>(h, ei, ei + E, dinv, acc, E);
    k_act<<<gF, 256, 0, stream>>>(acc, out, total, relu);
  };

  layer(x,    W1, b1, ei1, E1, dinv1, a,     0);  // h1 = elu(gcn1(x))
  layer(a,    W2, b2, ei2, E2, dinv2, zout,  0);  // z  = elu(gcn2(h1))  -> d_out[0:N*128]
  layer(zout, W3, b3, ei1, E1, dinv1, a,     0);  // h3 = elu(gcn3(z))
  layer(a,    W4, b4, ei2, E2, dinv2, xrout, 1);  // xr = relu(gcn4(h3)) -> d_out[N*128:]
}